// RBF_10651518894978
// MI455X (gfx1250) — compile-verified
//
#include <hip/hip_runtime.h>

typedef float v2f __attribute__((ext_vector_type(2)));
typedef float v8f __attribute__((ext_vector_type(8)));

constexpr int N1 = 8192, N2 = 8192, D = 8;
constexpr int TILE     = 16;
constexpr int TJ       = 8;                   // j-tiles per wave (A reuse)
constexpr int TILES_I  = N1 / TILE;           // 512
constexpr int TILES_J  = N2 / TILE;           // 512
constexpr int JGROUPS  = TILES_J / TJ;        // 64
constexpr int WAVES    = TILES_I * JGROUPS;   // 32768
constexpr int BLOCK    = 256;                 // 8 wave32s
constexpr int GRID     = (WAVES * 32) / BLOCK;// 4096

__global__ __launch_bounds__(BLOCK)
void RBF_10651518894978_kernel(const float* __restrict__ in1,
                               const float* __restrict__ in2,
                               const float* __restrict__ gammap,
                               float* __restrict__ out)
{
    const int tid  = blockIdx.x * BLOCK + threadIdx.x;
    const int wave = tid >> 5;
    const int lane = threadIdx.x & 31;
    const int l16  = lane & 15;
    const bool lo  = (lane < 16);

    const int ti        = wave / JGROUPS;
    const int tg        = wave - ti * JGROUPS;
    const int row_base  = ti * TILE;
    const int col_base0 = tg * (TILE * TJ);

    const float gamma = gammap[0];

    // s1 for row (row_base + l16): 8 contiguous f32 = 2x b128 loads
    const float* r1 = in1 + (size_t)(row_base + l16) * D;
    float4 p0 = ((const float4*)r1)[0];
    float4 p1 = ((const float4*)r1)[1];
    float s1 = ((p0.x + p0.y) + (p0.z + p0.w)) + ((p1.x + p1.y) + (p1.z + p1.w));

    // A matrix (16x4): col K=0 = s1, col K=1 = 1, cols K=2,3 = 0.
    // Lanes 0-15 hold {K=0,K=1}; lanes 16-31 hold {K=2,K=3} -> zeros.
    v2f a;
    a.x = lo ? s1   : 0.0f;
    a.y = lo ? 1.0f : 0.0f;

    for (int t = 0; t < TJ; ++t) {
        const int col_base = col_base0 + t * TILE;

        // s2 for column (col_base + l16)
        const float* r2 = in2 + (size_t)(col_base + l16) * D;
        float4 q0 = ((const float4*)r2)[0];
        float4 q1 = ((const float4*)r2)[1];
        float s2 = ((q0.x + q0.y) + (q0.z + q0.w)) + ((q1.x + q1.y) + (q1.z + q1.w));

        // B matrix (4x16): row K=0 = 1, row K=1 = -s2, rows K=2,3 = 0.
        v2f b;
        b.x = lo ? 1.0f : 0.0f;
        b.y = lo ? -s2  : 0.0f;

        // D[i,j] = s1[i] - s2[j] via one rank-2 outer product per 16x16 tile.
        v8f c = {};
        c = __builtin_amdgcn_wmma_f32_16x16x4_f32(
                /*neg_a=*/false, a, /*neg_b=*/false, b,
                /*c_mod=*/(short)0, c, /*reuse_a=*/false, /*reuse_b=*/false);

        // D layout: VGPR r -> row (row_base + r) for lanes 0-15,
        //                     row (row_base + r + 8) for lanes 16-31; col = col_base + l16.
        float* op = out + (size_t)(row_base + (lo ? 0 : 8)) * N2 + (col_base + l16);
        #pragma unroll
        for (int r = 0; r < 8; ++r) {
            float dv  = c[r];
            float res = __expf(dv * dv) * gamma;
            __builtin_nontemporal_store(res, op + (size_t)r * N2);
        }
    }
}

extern "C" void kernel_launch(void* const* d_in, const int* in_sizes, int n_in,
                              void* d_out, int out_size, void* d_ws, size_t ws_size,
                              hipStream_t stream) {
    const float* in1   = (const float*)d_in[0];
    const float* in2   = (const float*)d_in[1];
    const float* gamma = (const float*)d_in[2];
    float* out         = (float*)d_out;
    (void)in_sizes; (void)n_in; (void)out_size; (void)d_ws; (void)ws_size;

    RBF_10651518894978_kernel<<<GRID, BLOCK, 0, stream>>>(in1, in2, gamma, out);
}